// Aggregation_14061722927492
// MI455X (gfx1250) — compile-verified
//
#include <hip/hip_runtime.h>
#include <stdint.h>

#define NB   8
#define CH   256
#define HH   56
#define WW   56
#define CW   32
#define SHC  8            // share = C / Cw
#define KK   9
#define LL   (HH*WW)
#define TH   14           // output rows per tile
#define TR   (TH+2)       // LDS rows incl. halo
#define TCOL 64           // padded row: image col c -> LDS col c+4 (16B aligned)
#define PADL 4
#define NTILE (HH/TH)     // 4
#define NGROUP ((TH*WW)/4) // 196 float4 pixel-groups per tile

__global__ __launch_bounds__(256)
void agg_3x3_async_kernel(const float* __restrict__ inp,
                          const float* __restrict__ wgt,
                          float* __restrict__ out)
{
    __shared__ float tile[SHC][TR][TCOL];   // 32 KB

    const int tid = threadIdx.x;
    int b = blockIdx.x;
    const int t = b % NTILE;  b /= NTILE;
    const int g = b % CW;     b /= CW;
    const int n = b;
    const int h0 = t * TH;

    // ---- halo zeroing: cells DISJOINT from async-written interior, so it
    // ---- can overlap the async copies (single barrier at the end)
    for (int i = tid; i < SHC*TR; i += 256) {
        const int s = i / TR, rr = i % TR;
        tile[s][rr][PADL-1]  = 0.0f;        // image col -1
        tile[s][rr][PADL+WW] = 0.0f;        // image col 56
    }
    if (h0 == 0)                             // top halo row out of image
        for (int i = tid; i < SHC*(WW+2); i += 256)
            tile[i/(WW+2)][0][PADL-1 + i%(WW+2)] = 0.0f;
    if (h0 + TH == HH)                       // bottom halo row out of image
        for (int i = tid; i < SHC*(WW+2); i += 256)
            tile[i/(WW+2)][TR-1][PADL-1 + i%(WW+2)] = 0.0f;

    // ---- async global->LDS b128 staging of the interior (7 xfers/thread) ----
    const uint32_t lds0 = (uint32_t)(uintptr_t)&tile[0][0][0];
    const float* src_base = inp + (((size_t)n*CH + (size_t)g*SHC) * HH) * WW;
    for (int i = tid; i < SHC*TR*(WW/4); i += 256) {   // 1792 vec4 transfers
        const int s  = i / (TR*(WW/4));
        const int r2 = i % (TR*(WW/4));
        const int rr = r2 / (WW/4);
        const int j  = r2 % (WW/4);
        const int h  = h0 + rr - 1;
        if (h >= 0 && h < HH) {              // EXEC-predicated async copy
            const uint64_t ga = (uint64_t)(uintptr_t)
                (src_base + ((size_t)s*HH + h)*WW + 4*j);       // 16B aligned
            const uint32_t la = lds0 +
                (uint32_t)(((s*TR + rr)*TCOL + PADL + 4*j) * 4); // 16B aligned
            asm volatile("global_load_async_to_lds_b128 %0, %1, off"
                         :: "v"(la), "v"(ga) : "memory");
        }
    }
    asm volatile("s_wait_asynccnt 0" ::: "memory");
    __syncthreads();                         // DScnt waited by compiler here

    // ---- compute: one group of 4 consecutive pixels per thread ----
    if (tid < NGROUP) {
        const int p0 = tid * 4;
        const int r  = p0 / WW;
        const int w0 = p0 % WW;              // multiple of 4
        const int l  = (h0 + r)*WW + w0;

        const float* wbase = wgt + (((size_t)n*CW + g) * KK) * LL + l;
        float*       obase = out + (((size_t)n*CH + (size_t)g*SHC) * LL) + l;

        float wk[KK][4];
#pragma unroll
        for (int k = 0; k < KK; ++k) {       // 9 coalesced b128 weight loads
            const float4 wv = *(const float4*)(wbase + (size_t)k*LL);
            wk[k][0]=wv.x; wk[k][1]=wv.y; wk[k][2]=wv.z; wk[k][3]=wv.w;
        }

#pragma unroll
        for (int s = 0; s < SHC; ++s) {
            float acc[4] = {0.0f, 0.0f, 0.0f, 0.0f};
#pragma unroll
            for (int i2 = 0; i2 < 3; ++i2) {
                float row6[6];               // 6-wide window shared by 4 pixels
#pragma unroll
                for (int q = 0; q < 6; ++q)
                    row6[q] = tile[s][r + i2][w0 + (PADL-1) + q];
#pragma unroll
                for (int j = 0; j < 3; ++j)
#pragma unroll
                    for (int px = 0; px < 4; ++px)
                        acc[px] = fmaf(wk[i2*3 + j][px], row6[px + j], acc[px]);
            }
            float4 ov; ov.x=acc[0]; ov.y=acc[1]; ov.z=acc[2]; ov.w=acc[3];
            *(float4*)(obase + (size_t)s*LL) = ov;   // coalesced b128 store
        }
    }
}

extern "C" void kernel_launch(void* const* d_in, const int* in_sizes, int n_in,
                              void* d_out, int out_size, void* d_ws, size_t ws_size,
                              hipStream_t stream) {
    const float* inp = (const float*)d_in[0];   // [8,256,56,56] f32
    const float* wgt = (const float*)d_in[1];   // [8,32,9,3136] f32
    float* out = (float*)d_out;                 // [8,256,56,56] f32
    dim3 grid(NB * CW * NTILE);                 // 1024 blocks, 8 waves each
    agg_3x3_async_kernel<<<grid, 256, 0, stream>>>(inp, wgt, out);
}